// IFNode5MaskD_49134425866322
// MI455X (gfx1250) — compile-verified
//
#include <hip/hip_runtime.h>

// D[32, N] = heaviside( (W .* mask)[32,32] @ X[32, N] + bias ), N = 2^20.
// Memory-bound: ~268 MB of traffic @ 23.3 TB/s => ~11.5 us floor. Staying
// BW-bound needs ~186 TF/s of f32, above VALU peak => f32 WMMA pipe
// (V_WMMA_F32_16X16X4_F32). All element indices fit in u32 (<= 32M elems,
// 128 MB) => pure 32-bit running offsets so every global access lowers to
// saddr + scaled-u32-voffset (no 64-bit VALU address math).
// NOTE: no matrix_a_reuse hints — the ISA only permits them when the next
// instruction is identical, and the scheduler reorders WMMAs, which would
// make the hint undefined-behavior on hardware.

#define T_DIM 32

typedef __attribute__((ext_vector_type(2)))  float v2f;
typedef __attribute__((ext_vector_type(8)))  float v8f;

// mask = 0.5*tril + 0.5 => 1.0 for tp <= t, 0.5 above the diagonal.
// Loads the (even-aligned) K-pair as one b64 and applies the mask.
__device__ __forceinline__ v2f load_a(const float* __restrict__ w, int t, int ka) {
  v2f wv;
  wv.x = w[t * T_DIM + ka];
  wv.y = w[t * T_DIM + ka + 1];
  wv.x *= (ka     <= t) ? 1.0f : 0.5f;
  wv.y *= (ka + 1 <= t) ? 1.0f : 0.5f;
  return wv;
}

__global__ __launch_bounds__(256) void
spike_linear_wmma(const float* __restrict__ x, const float* __restrict__ w,
                  const float* __restrict__ bias, float* __restrict__ out,
                  unsigned N) {
  const int lane  = threadIdx.x & 31;
  const int wave  = threadIdx.x >> 5;
  const int col   = lane & 15;   // column within a 16-wide tile / A-row index
  const int khalf = lane >> 4;   // 0: lanes 0-15, 1: lanes 16-31

  // Each wave: 32 rows x 32 cols (two 16-col tiles sharing A fragments).
  // 8 waves/block => 256 cols/block.
  unsigned nb = blockIdx.x * 256u + (unsigned)wave * 32u;
  if (nb + 32u > N) nb = N - 32u;  // uniform per-wave clamp (EXEC stays all-1s)

  // ---- Preload all A fragments (masked weights) once: 8 k-steps x 2 M-tiles.
  // A (16x4 f32 layout): lane holds row = col (or col+16), K = ka, ka+1 where
  // ka = 4*i + 2*khalf.
  v2f a0[8], a1[8];
#pragma unroll
  for (int i = 0; i < 8; ++i) {
    const int ka = i * 4 + khalf * 2;
    a0[i] = load_a(w, col,      ka);
    a1[i] = load_a(w, col + 16, ka);
  }

  v8f acc00 = {};  // rows  0..15, cols nb..nb+15
  v8f acc01 = {};  // rows  0..15, cols nb+16..nb+31
  v8f acc10 = {};  // rows 16..31, cols nb..nb+15
  v8f acc11 = {};  // rows 16..31, cols nb+16..nb+31

  // ---- K = 32 as 8 chained 16x16x4 f32 WMMAs per accumulator.
  // B (4x16): lane holds column = col, K-pair = ka, ka+1. Four independent u32
  // running element indices (rows ka / ka+1, col tiles +0 / +16), each bumped
  // by 4*N per k-step => every load is saddr + scale_offset register form.
  unsigned oA = (unsigned)(khalf * 2) * N + nb + (unsigned)col;  // row ka,   +0
  unsigned oB = oA + N;                                          // row ka+1, +0
  unsigned oC = oA + 16u;                                        // row ka,   +16
  unsigned oD = oB + 16u;                                        // row ka+1, +16
  const unsigned step = 4u * N;
#pragma unroll
  for (int i = 0; i < 8; ++i) {
    v2f b0, b1;
    b0.x = __builtin_nontemporal_load(&x[oA]);
    b0.y = __builtin_nontemporal_load(&x[oB]);
    b1.x = __builtin_nontemporal_load(&x[oC]);
    b1.y = __builtin_nontemporal_load(&x[oD]);
    acc00 = __builtin_amdgcn_wmma_f32_16x16x4_f32(false, a0[i], false, b0,
                                                  (short)0, acc00, false, false);
    acc01 = __builtin_amdgcn_wmma_f32_16x16x4_f32(false, a0[i], false, b1,
                                                  (short)0, acc01, false, false);
    acc10 = __builtin_amdgcn_wmma_f32_16x16x4_f32(false, a1[i], false, b0,
                                                  (short)0, acc10, false, false);
    acc11 = __builtin_amdgcn_wmma_f32_16x16x4_f32(false, a1[i], false, b1,
                                                  (short)0, acc11, false, false);
    oA += step;
    oB += step;
    oC += step;
    oD += step;
  }

  // ---- Epilogue. C/D 16x16 f32 layout: lane holds column = col; VGPR v holds
  // row v + 8*khalf. Bias add, Heaviside, streaming stores with u32 running
  // row offsets (+N per row).
  const int rbase = khalf * 8;
  unsigned ro0 = (unsigned)rbase * N + nb + (unsigned)col;  // rows rbase+v,    +0
  unsigned ro1 = ro0 + 16u * N;                             // rows 16+rbase+v, +0
  unsigned ro2 = ro0 + 16u;                                 // rows rbase+v,    +16
  unsigned ro3 = ro1 + 16u;                                 // rows 16+rbase+v, +16
#pragma unroll
  for (int v = 0; v < 8; ++v) {
    const float bl = bias[rbase + v];
    const float bh = bias[16 + rbase + v];
    __builtin_nontemporal_store(acc00[v] + bl >= 0.0f ? 1.0f : 0.0f, &out[ro0]);
    __builtin_nontemporal_store(acc01[v] + bl >= 0.0f ? 1.0f : 0.0f, &out[ro2]);
    __builtin_nontemporal_store(acc10[v] + bh >= 0.0f ? 1.0f : 0.0f, &out[ro1]);
    __builtin_nontemporal_store(acc11[v] + bh >= 0.0f ? 1.0f : 0.0f, &out[ro3]);
    ro0 += N;
    ro1 += N;
    ro2 += N;
    ro3 += N;
  }
}

extern "C" void kernel_launch(void* const* d_in, const int* in_sizes, int n_in,
                              void* d_out, int out_size, void* d_ws, size_t ws_size,
                              hipStream_t stream) {
  const float* x    = (const float*)d_in[0];  // [32, N] flattened (T,B,C,H,W)
  const float* w    = (const float*)d_in[1];  // [32, 32]
  const float* bias = (const float*)d_in[2];  // [32]
  float* out = (float*)d_out;

  const unsigned N = (unsigned)(in_sizes[0] / T_DIM);  // 1048576
  const unsigned colsPerBlock = 256;                   // 8 waves * 32 cols
  const unsigned grid = (N + colsPerBlock - 1) / colsPerBlock;

  spike_linear_wmma<<<grid, 256, 0, stream>>>(x, w, bias, out, N);
}